// AFT_Full_55301998903700
// MI455X (gfx1250) — compile-verified
//
#include <hip/hip_runtime.h>
#include <hip/hip_bf16.h>
#include <math.h>
#include <stdint.h>

typedef __attribute__((ext_vector_type(2))) float v2f;
typedef __attribute__((ext_vector_type(8))) float v8f;

#define NB   16
#define NC   64
#define NH   128
#define NW   128
#define NHID 128
#define XPAD 66   // padded row stride (floats) for x tiles in LDS
#define YPAD 18   // padded row stride (floats) for per-wave y scratch

static __device__ __forceinline__ v8f wmma_f32_k4(v2f a, v2f b, v8f c) {
  // D = A(16x4 f32) * B(4x16 f32) + C(16x16 f32)  -> v_wmma_f32_16x16x4_f32
  return __builtin_amdgcn_wmma_f32_16x16x4_f32(false, a, false, b, (short)0, c,
                                               false, false);
}

// ---------------------------------------------------------------------------
// Pass 1: m[h,w,d] = max_b ( X[b,h,w,:] . wk[d,:] )      (bk cancels in exp)
// grid = H * (W/16) = 1024 blocks, 256 threads (8 waves = 8 d-tiles of 16)
// Kept on the global_load+ds_store path: the compiler software-pipelines the
// next batch's loads under the current batch's WMMAs (verified in round-1 asm).
// ---------------------------------------------------------------------------
__global__ __launch_bounds__(256) void aft_kmax(const float* __restrict__ x,
                                                const float* __restrict__ wk,
                                                float* __restrict__ m) {
  __shared__ float sx[16 * XPAD];

  const int tid  = threadIdx.x;
  const int lane = tid & 31;
  const int wave = tid >> 5;
  const int h    = blockIdx.x >> 3;
  const int w0   = (blockIdx.x & 7) << 4;
  const int n0   = wave << 4;
  const int nl   = lane & 15;          // N (or M) index within tile
  const int khi  = (lane >> 4) << 1;   // lanes 16-31 hold K+2,K+3
  const int Mb   = (lane >> 4) << 3;   // D-layout M base (0 or 8)

  // Preload B fragments of wk for this wave's d-tile: B[K=c][N=d] = wk[d*NC+c]
  v2f bf[16];
#pragma unroll
  for (int s = 0; s < 16; ++s) {
    const float* wr = wk + (size_t)(n0 + nl) * NC + 4 * s + khi;
    bf[s].x = wr[0];
    bf[s].y = wr[1];
  }

  v8f vmax = -3.402823466e38f;

  for (int b = 0; b < NB; ++b) {
    __syncthreads();
    {  // stage X tile [16 w x 64 c] cooperatively (coalesced float4)
      const int row = tid >> 4;
      const int c4  = (tid & 15) << 2;
      const float4 t =
          *(const float4*)(x + ((size_t)((b * NH + h) * NW + w0 + row)) * NC + c4);
      float* dst = &sx[row * XPAD + c4];
      dst[0] = t.x; dst[1] = t.y; dst[2] = t.z; dst[3] = t.w;
    }
    __syncthreads();

    v8f acc = 0.0f;
#pragma unroll
    for (int s = 0; s < 16; ++s) {
      v2f a;
      const float* ar = &sx[nl * XPAD + 4 * s + khi];
      a.x = ar[0];
      a.y = ar[1];
      acc = wmma_f32_k4(a, bf[s], acc);
    }
#pragma unroll
    for (int r = 0; r < 8; ++r) vmax[r] = fmaxf(vmax[r], acc[r]);
  }

#pragma unroll
  for (int r = 0; r < 8; ++r)
    m[((size_t)(h * NW + w0 + Mb + r)) * NHID + n0 + nl] = vmax[r];
}

// ---------------------------------------------------------------------------
// Pass 2: per (b,h) row block.
//   stage X via GLOBAL_LOAD_ASYNC_TO_LDS_B128 (ASYNCcnt path, no VGPR bounce)
//   K,V tiles -> S_num/S_den column sums (LDS ds_add_f32)
//   Q tiles   -> y = sigmoid(q+bq) * S_num/S_den
//   out       -> y @ wo^T + bo,  stored flat [B,H,W,C]
// grid = B*H = 2048 blocks, 256 threads (wave i owns w-tile i)
// ---------------------------------------------------------------------------
__global__ __launch_bounds__(256) void aft_main(
    const float* __restrict__ x,
    const float* __restrict__ wq, const float* __restrict__ bq,
    const float* __restrict__ wk,
    const float* __restrict__ wv, const float* __restrict__ bv,
    const float* __restrict__ wo, const float* __restrict__ bo,
    const float* __restrict__ m, float* __restrict__ out) {
  __shared__ float sx[NW * XPAD];       // 33.0 KB: full [128w x 64c] block
  __shared__ float sy[8 * 16 * YPAD];   //  9.0 KB: per-wave y transpose scratch
  __shared__ float sn[NHID];            // S_num
  __shared__ float sd[NHID];            // S_den

  const int tid  = threadIdx.x;
  const int lane = tid & 31;
  const int wave = tid >> 5;
  const int p    = blockIdx.x;          // b*NH + h
  const int h    = p & (NH - 1);
  const int w0   = wave << 4;
  const int nl   = lane & 15;
  const int khi  = (lane >> 4) << 1;
  const int Mb   = (lane >> 4) << 3;

  {  // stage X [128 x 64] straight into LDS via the async path (16B per lane,
     // per-lane LDS dest keeps the padded stride-66 conflict-free layout)
    const float* xb = x + (size_t)p * NW * NC;
#pragma unroll
    for (int it = 0; it < 8; ++it) {
      const int row = (it << 4) + (tid >> 4);
      const int c4  = (tid & 15) << 2;
      const unsigned lds =
          (unsigned)(uintptr_t)(&sx[row * XPAD + c4]);
      const unsigned long long ga =
          (unsigned long long)(uintptr_t)(xb + (size_t)row * NC + c4);
      asm volatile("global_load_async_to_lds_b128 %0, %1, off"
                   :: "v"(lds), "v"(ga)
                   : "memory");
    }
  }
  if (tid < NHID) sn[tid] = 0.0f;
  else            sd[tid - NHID] = 0.0f;
  asm volatile("s_wait_asynccnt 0x0" ::: "memory");
  __syncthreads();

  // ------------------ step 2: K,V -> S reductions ------------------
  for (int j = 0; j < 8; ++j) {
    const int d0 = j << 4;

    v8f kacc = 0.0f;
    {
      v2f bf[16];
#pragma unroll
      for (int s = 0; s < 16; ++s) {
        const float* wr = wk + (size_t)(d0 + nl) * NC + 4 * s + khi;
        bf[s].x = wr[0]; bf[s].y = wr[1];
      }
#pragma unroll
      for (int s = 0; s < 16; ++s) {
        v2f a;
        const float* ar = &sx[(w0 + nl) * XPAD + 4 * s + khi];
        a.x = ar[0]; a.y = ar[1];
        kacc = wmma_f32_k4(a, bf[s], kacc);
      }
    }
    v8f vacc = 0.0f;
    {
      v2f bf[16];
#pragma unroll
      for (int s = 0; s < 16; ++s) {
        const float* wr = wv + (size_t)(d0 + nl) * NC + 4 * s + khi;
        bf[s].x = wr[0]; bf[s].y = wr[1];
      }
#pragma unroll
      for (int s = 0; s < 16; ++s) {
        v2f a;
        const float* ar = &sx[(w0 + nl) * XPAD + 4 * s + khi];
        a.x = ar[0]; a.y = ar[1];
        vacc = wmma_f32_k4(a, bf[s], vacc);
      }
    }

    const float  bvn  = bv[d0 + nl];
    const float* mrow = m + ((size_t)(h * NW + w0 + Mb)) * NHID + d0 + nl;
    float pnum = 0.0f, pden = 0.0f;
#pragma unroll
    for (int r = 0; r < 8; ++r) {
      const float e = __expf(kacc[r] - mrow[(size_t)r * NHID]);
      pden += e;
      pnum += e * (vacc[r] + bvn);
    }
    atomicAdd(&sn[d0 + nl], pnum);   // ds_add_f32 (both lane halves, same d)
    atomicAdd(&sd[d0 + nl], pden);
  }
  __syncthreads();

  // ------------------ step 3: Q -> y -> out ------------------
  v8f oacc[4];
#pragma unroll
  for (int ct = 0; ct < 4; ++ct) oacc[ct] = 0.0f;

  float* syw = &sy[wave * 16 * YPAD];

  for (int j = 0; j < 8; ++j) {
    const int d0 = j << 4;

    v8f qacc = 0.0f;
    {
      v2f bf[16];
#pragma unroll
      for (int s = 0; s < 16; ++s) {
        const float* wr = wq + (size_t)(d0 + nl) * NC + 4 * s + khi;
        bf[s].x = wr[0]; bf[s].y = wr[1];
      }
#pragma unroll
      for (int s = 0; s < 16; ++s) {
        v2f a;
        const float* ar = &sx[(w0 + nl) * XPAD + 4 * s + khi];
        a.x = ar[0]; a.y = ar[1];
        qacc = wmma_f32_k4(a, bf[s], qacc);
      }
    }

    const float bqn   = bq[d0 + nl];
    const float ratio = sn[d0 + nl] / sd[d0 + nl];

    // y = sigmoid(q)*ratio, dump D-layout tile to per-wave scratch
#pragma unroll
    for (int r = 0; r < 8; ++r) {
      const float qv = qacc[r] + bqn;
      syw[(Mb + r) * YPAD + nl] = ratio / (1.0f + __expf(-qv));
    }
    // same-wave LDS RAW: ordered by dscnt waits, no barrier needed.

    // out partial: A = y (16w x 16d slab), B[K=d][N=c] = wo[c*NHID + d]
#pragma unroll
    for (int s = 0; s < 4; ++s) {
      v2f a;
      const float* ar = &syw[nl * YPAD + 4 * s + khi];
      a.x = ar[0]; a.y = ar[1];
#pragma unroll
      for (int ct = 0; ct < 4; ++ct) {
        v2f bfr;
        const float* wr = wo + (size_t)(ct * 16 + nl) * NHID + d0 + 4 * s + khi;
        bfr.x = wr[0]; bfr.y = wr[1];
        oacc[ct] = wmma_f32_k4(a, bfr, oacc[ct]);
      }
    }
  }

  float* ob = out + (size_t)p * NW * NC;
#pragma unroll
  for (int ct = 0; ct < 4; ++ct) {
    const float bon = bo[ct * 16 + nl];
#pragma unroll
    for (int r = 0; r < 8; ++r)
      ob[(size_t)(w0 + Mb + r) * NC + ct * 16 + nl] = oacc[ct][r] + bon;
  }
}

// ---------------------------------------------------------------------------
extern "C" void kernel_launch(void* const* d_in, const int* in_sizes, int n_in,
                              void* d_out, int out_size, void* d_ws,
                              size_t ws_size, hipStream_t stream) {
  (void)in_sizes; (void)n_in; (void)out_size; (void)ws_size;
  const float* x  = (const float*)d_in[0];
  const float* wq = (const float*)d_in[1];
  const float* bq = (const float*)d_in[2];
  const float* wk = (const float*)d_in[3];
  // d_in[4] = bk : unused — cancels exactly in exp(k - max_b k)
  const float* wv = (const float*)d_in[5];
  const float* bv = (const float*)d_in[6];
  // d_in[7] = w  : unused — exp(w - max over size-1 axis) == 1 exactly
  const float* wo = (const float*)d_in[8];
  const float* bo = (const float*)d_in[9];
  float* out = (float*)d_out;
  float* m   = (float*)d_ws;  // [H, W, HID] fp32 = 8.39 MB scratch

  aft_kmax<<<NH * (NW / 16), 256, 0, stream>>>(x, wk, m);
  aft_main<<<NB * NH, 256, 0, stream>>>(x, wq, bq, wk, wv, bv, wo, bo, m, out);
}